// BaselineSALayer_11596411699409
// MI455X (gfx1250) — compile-verified
//
#include <hip/hip_runtime.h>
#include <hip/hip_bf16.h>

// ---------------------------------------------------------------------------
// PointNet++ SA layer for MI455X (gfx1250, wave32).
// GEMMs use V_WMMA_F32_16X16X4_F32 (fp32 WMMA; the layer is HBM-bound at
// ~270MB traffic per GEMM vs 23.3 TB/s, so fp32 keeps accuracy for free).
// Requires ws_size >= ~566 MB (see offsets below).
// ---------------------------------------------------------------------------

typedef __attribute__((ext_vector_type(2))) float v2f;
typedef __attribute__((ext_vector_type(8))) float v8f;

#define BATCH 16
#define NPTS  4096
#define SPTS  1024
#define KNN   32
#define CIN0  64
#define MROWS (BATCH * SPTS * KNN)   // 524288

// ---- workspace offsets (bytes) --------------------------------------------
static constexpr size_t OFF_XYZT = 0x00000000;  // [B,N,3]   f32   786432
static constexpr size_t OFF_PTST = 0x000C0000;  // [B,N,64]  f32 16777216
static constexpr size_t OFF_CIDX = 0x010C0000;  // [B,S]     i32    65536
static constexpr size_t OFF_NEWX = 0x010D0000;  // [B,S,3]   f32   196608
static constexpr size_t OFF_GIDX = 0x01100000;  // [B,S,K]   i32  2097152
static constexpr size_t OFF_FEAT = 0x01300000;  // [M,68] f32 (reused as Y1 [M,64])
static constexpr size_t OFF_Y0   = 0x09B00000;  // [M,64]    f32 134217728
static constexpr size_t OFF_Y2   = 0x11B00000;  // [M,128]   f32 268435456
static constexpr size_t OFF_PART = 0x21B00000;  // 2*256*128 f32   262144
static constexpr size_t OFF_SCL  = 0x21B40000;  // 128 f32
static constexpr size_t OFF_SHF  = 0x21B40200;  // 128 f32

// ---------------------------------------------------------------------------
// 0) transpose xyz [B,3,N]->[B,N,3] and points [B,64,N]->[B,N,64]
// ---------------------------------------------------------------------------
__global__ void k_transpose(const float* __restrict__ xyz,
                            const float* __restrict__ pts,
                            float* __restrict__ xyz_t,
                            float* __restrict__ pts_t) {
  const size_t stride = (size_t)gridDim.x * blockDim.x;
  const size_t t1 = (size_t)BATCH * NPTS * 3;
  for (size_t o = (size_t)blockIdx.x * blockDim.x + threadIdx.x; o < t1; o += stride) {
    size_t d = o % 3, n = (o / 3) % NPTS, b = o / (3 * NPTS);
    xyz_t[o] = xyz[(b * 3 + d) * NPTS + n];
  }
  const size_t t2 = (size_t)BATCH * NPTS * CIN0;
  for (size_t o = (size_t)blockIdx.x * blockDim.x + threadIdx.x; o < t2; o += stride) {
    size_t c = o % CIN0, n = (o / CIN0) % NPTS, b = o / ((size_t)CIN0 * NPTS);
    pts_t[o] = pts[(b * CIN0 + c) * NPTS + n];
  }
}

// ---------------------------------------------------------------------------
// 1) farthest point sampling: one block per batch, 256 threads, 16 pts/thread
// ---------------------------------------------------------------------------
__global__ void __launch_bounds__(256) k_fps(const float* __restrict__ xyz_t,
                                             int* __restrict__ cidx,
                                             float* __restrict__ new_xyz) {
  const int b = blockIdx.x;
  const int t = threadIdx.x;
  const float* P = xyz_t + (size_t)b * NPTS * 3;

  float px[16], py[16], pz[16], dist[16];
#pragma unroll
  for (int i = 0; i < 16; ++i) {
    int n = t + i * 256;
    px[i] = P[n * 3 + 0]; py[i] = P[n * 3 + 1]; pz[i] = P[n * 3 + 2];
    dist[i] = 1e10f;
  }

  __shared__ float s_c[3];
  __shared__ float s_val[8];
  __shared__ int   s_idx[8];
  __shared__ int   s_far;

  int farthest = 0;
  for (int it = 0; it < SPTS; ++it) {
    if (t == 0) {
      cidx[b * SPTS + it] = farthest;
      float cx = P[farthest * 3 + 0];
      float cy = P[farthest * 3 + 1];
      float cz = P[farthest * 3 + 2];
      s_c[0] = cx; s_c[1] = cy; s_c[2] = cz;
      float* nx = new_xyz + ((size_t)b * SPTS + it) * 3;
      nx[0] = cx; nx[1] = cy; nx[2] = cz;
    }
    __syncthreads();
    const float cx = s_c[0], cy = s_c[1], cz = s_c[2];
    float best = -1.0f; int bidx = 0x7fffffff;
#pragma unroll
    for (int i = 0; i < 16; ++i) {
      float dx = px[i] - cx, dy = py[i] - cy, dz = pz[i] - cz;
      float d = dx * dx + dy * dy + dz * dz;
      dist[i] = fminf(dist[i], d);
      int n = t + i * 256;
      if (dist[i] > best || (dist[i] == best && n < bidx)) { best = dist[i]; bidx = n; }
    }
    // wave32 reduction (argmax, lowest index wins ties)
#pragma unroll
    for (int off = 16; off > 0; off >>= 1) {
      float ov = __shfl_xor(best, off, 32);
      int   oi = __shfl_xor(bidx, off, 32);
      if (ov > best || (ov == best && oi < bidx)) { best = ov; bidx = oi; }
    }
    const int lane = t & 31, w = t >> 5;
    if (lane == 0) { s_val[w] = best; s_idx[w] = bidx; }
    __syncthreads();
    if (t == 0) {
      float bv = s_val[0]; int bi = s_idx[0];
      for (int j = 1; j < 8; ++j)
        if (s_val[j] > bv || (s_val[j] == bv && s_idx[j] < bi)) { bv = s_val[j]; bi = s_idx[j]; }
      s_far = bi;
    }
    __syncthreads();
    farthest = s_far;
  }
}

// ---------------------------------------------------------------------------
// 2) kNN: block = 256 centroids of one batch; cloud staged in LDS (48KB);
//    32-entry register-resident sorted insertion (fully unrolled).
// ---------------------------------------------------------------------------
__global__ void __launch_bounds__(256) k_knn(const float* __restrict__ xyz_t,
                                             const float* __restrict__ new_xyz,
                                             int* __restrict__ gidx) {
  __shared__ float sx[NPTS], sy[NPTS], sz[NPTS];
  const int b = blockIdx.x >> 2;
  const int chunk = blockIdx.x & 3;
  const int s = chunk * 256 + threadIdx.x;
  const float* P = xyz_t + (size_t)b * NPTS * 3;
  for (int i = threadIdx.x; i < NPTS; i += 256) {
    sx[i] = P[i * 3 + 0]; sy[i] = P[i * 3 + 1]; sz[i] = P[i * 3 + 2];
  }
  __syncthreads();

  const float* cen = new_xyz + ((size_t)b * SPTS + s) * 3;
  const float cx = cen[0], cy = cen[1], cz = cen[2];

  float kd[KNN]; int ki[KNN];
#pragma unroll
  for (int j = 0; j < KNN; ++j) { kd[j] = 3.4e38f; ki[j] = 0; }

  for (int n = 0; n < NPTS; ++n) {
    float dx = sx[n] - cx, dy = sy[n] - cy, dz = sz[n] - cz;
    float d = dx * dx + dy * dy + dz * dz;
    if (d < kd[KNN - 1]) {
      int pos = 0;
#pragma unroll
      for (int j = 0; j < KNN; ++j) pos += (kd[j] <= d) ? 1 : 0;
#pragma unroll
      for (int j = KNN - 1; j >= 1; --j) {
        if (kd[j - 1] > d) { kd[j] = kd[j - 1]; ki[j] = ki[j - 1]; }
      }
#pragma unroll
      for (int j = 0; j < KNN; ++j) if (j == pos) { kd[j] = d; ki[j] = n; }
    }
  }
  int* g = gidx + ((size_t)b * SPTS + s) * KNN;
#pragma unroll
  for (int j = 0; j < KNN; ++j) g[j] = ki[j];
}

// ---------------------------------------------------------------------------
// 3) build feature rows [M, 68]: cols 0..2 = grouped_xyz - centroid,
//    3..66 = gathered point feats, 67 = zero pad. One wave per row.
// ---------------------------------------------------------------------------
__global__ void __launch_bounds__(256) k_build_feat(const float* __restrict__ xyz_t,
                                                    const float* __restrict__ pts_t,
                                                    const float* __restrict__ new_xyz,
                                                    const int* __restrict__ gidx,
                                                    float* __restrict__ feat) {
  const int gw = (int)((blockIdx.x * 256 + threadIdx.x) >> 5);  // row, exact grid
  const int lane = threadIdx.x & 31;
  const int s = (gw >> 5) & (SPTS - 1);
  const int b = gw >> 15;
  const int idx = gidx[gw];
  const float* p  = pts_t + ((size_t)b * NPTS + idx) * CIN0;
  const float* px = xyz_t + ((size_t)b * NPTS + idx) * 3;
  const float* cen = new_xyz + ((size_t)b * SPTS + s) * 3;
  float* f = feat + (size_t)gw * 68;

  // col = lane
  float v0 = (lane < 3) ? (px[lane] - cen[lane]) : p[lane - 3];
  f[lane] = v0;
  // col = lane + 32
  f[lane + 32] = p[lane + 29];
  // cols 64..67
  if (lane < 4) f[64 + lane] = (lane < 3) ? p[61 + lane] : 0.0f;
}

// ---------------------------------------------------------------------------
// 4) GEMM with V_WMMA_F32_16X16X4_F32: Y[M,COUT] = A[M,KD] * W[COUT,CIN]^T + b
//    8 waves/block, each wave owns a 16-row tile x all COUT cols.
// ---------------------------------------------------------------------------
template <int KD, int CINR, int COUT>
__global__ void __launch_bounds__(256) k_gemm_wmma(const float* __restrict__ A,
                                                   const float* __restrict__ W,
                                                   const float* __restrict__ bias,
                                                   float* __restrict__ Y) {
  __shared__ float sW[COUT * KD];
  __shared__ float sB[COUT];
  for (int i = threadIdx.x; i < COUT * KD; i += 256) {
    int n = i / KD, k = i % KD;
    sW[i] = (k < CINR) ? W[n * CINR + k] : 0.0f;
  }
  for (int i = threadIdx.x; i < COUT; i += 256) sB[i] = bias[i];
  __syncthreads();

  const int wave = threadIdx.x >> 5;
  const int lane = threadIdx.x & 31;
  const int lr = lane & 15;          // row-in-tile for A, col for B/D
  const int kh = (lane >> 4) * 2;    // K sub-offset 0 or 2
  const size_t m0 = ((size_t)blockIdx.x * 8 + wave) * 16;

  constexpr int NT = COUT / 16;
  v8f acc[NT];
#pragma unroll
  for (int t = 0; t < NT; ++t) acc[t] = (v8f){0.f,0.f,0.f,0.f,0.f,0.f,0.f,0.f};

  const float* ap = A + (m0 + lr) * KD + kh;
  for (int k0 = 0; k0 < KD; k0 += 4) {
    v2f a = *(const v2f*)(ap + k0);
#pragma unroll
    for (int t = 0; t < NT; ++t) {
      const float* wp = &sW[(t * 16 + lr) * KD + kh + k0];
      v2f bm = { wp[0], wp[1] };
      acc[t] = __builtin_amdgcn_wmma_f32_16x16x4_f32(
          false, a, false, bm, (short)0, acc[t], false, false);
    }
  }

  const int rh = (lane >> 4) * 8;  // D: VGPR j -> row j + 8*(lane>>4), col lane&15
#pragma unroll
  for (int t = 0; t < NT; ++t) {
    const float bv = sB[t * 16 + lr];
#pragma unroll
    for (int j = 0; j < 8; ++j) {
      Y[(m0 + rh + j) * COUT + t * 16 + lr] = acc[t][j] + bv;
    }
  }
}

// ---------------------------------------------------------------------------
// 5) BatchNorm (training stats), deterministic two-stage reduction
// ---------------------------------------------------------------------------
template <int COUT>
__global__ void __launch_bounds__(256) k_bn_partial(const float* __restrict__ Y,
                                                    float* __restrict__ psum,
                                                    float* __restrict__ psq) {
  const size_t total = (size_t)MROWS * COUT;
  const size_t tidg = (size_t)blockIdx.x * 256 + threadIdx.x;  // grid fixed at 256 blocks
  float s = 0.f, s2 = 0.f;
  for (size_t e = tidg; e < total; e += 65536) { float v = Y[e]; s += v; s2 += v * v; }
  __shared__ float ls[256], ls2[256];
  ls[threadIdx.x] = s; ls2[threadIdx.x] = s2;
  __syncthreads();
  if (threadIdx.x < COUT) {
    float a = 0.f, a2 = 0.f;
    for (int i = threadIdx.x; i < 256; i += COUT) { a += ls[i]; a2 += ls2[i]; }
    psum[blockIdx.x * COUT + threadIdx.x] = a;
    psq [blockIdx.x * COUT + threadIdx.x] = a2;
  }
}

template <int COUT>
__global__ void __launch_bounds__(256) k_bn_finalize(const float* __restrict__ psum,
                                                     const float* __restrict__ psq,
                                                     const float* __restrict__ g,
                                                     const float* __restrict__ beta,
                                                     float* __restrict__ scale,
                                                     float* __restrict__ shift) {
  const int c = threadIdx.x;
  if (c < COUT) {
    float s = 0.f, s2 = 0.f;
    for (int i = 0; i < 256; ++i) { s += psum[i * COUT + c]; s2 += psq[i * COUT + c]; }
    const float invM = 1.0f / (float)MROWS;
    float mu = s * invM;
    float var = s2 * invM - mu * mu;
    float sc = g[c] * rsqrtf(var + 1e-5f);
    scale[c] = sc;
    shift[c] = beta[c] - mu * sc;
  }
}

template <int COUT>
__global__ void __launch_bounds__(256) k_bn_apply(float* __restrict__ Y,
                                                  const float* __restrict__ scale,
                                                  const float* __restrict__ shift) {
  const size_t total = (size_t)MROWS * COUT;
  const size_t stride = (size_t)gridDim.x * blockDim.x;
  for (size_t e = (size_t)blockIdx.x * blockDim.x + threadIdx.x; e < total; e += stride) {
    int c = (int)(e % COUT);
    Y[e] = fmaxf(0.0f, Y[e] * scale[c] + shift[c]);
  }
}

// ---------------------------------------------------------------------------
// 6) max over K=32 + transpose -> out2[b,c,s]; one wave per (b,s)
// ---------------------------------------------------------------------------
__global__ void __launch_bounds__(256) k_maxpool_tr(const float* __restrict__ Y2,
                                                    float* __restrict__ out2) {
  const int gw = (int)((blockIdx.x * 256 + threadIdx.x) >> 5);  // b*S + s
  const int lane = threadIdx.x & 31;
  const int s = gw & (SPTS - 1);
  const int b = gw >> 10;
  const float* base = Y2 + (size_t)gw * KNN * 128;
  float m[4] = {-3.4e38f, -3.4e38f, -3.4e38f, -3.4e38f};
  for (int k = 0; k < KNN; ++k) {
#pragma unroll
    for (int j = 0; j < 4; ++j)
      m[j] = fmaxf(m[j], base[k * 128 + lane + 32 * j]);
  }
#pragma unroll
  for (int j = 0; j < 4; ++j)
    out2[((size_t)b * 128 + lane + 32 * j) * SPTS + s] = m[j];
}

// out1[b,d,s] = new_xyz[b,s,d]
__global__ void k_newxyz_out(const float* __restrict__ new_xyz, float* __restrict__ out1) {
  const int o = blockIdx.x * 256 + threadIdx.x;  // exact: 49152
  const int s = o & (SPTS - 1);
  const int d = (o >> 10) % 3;
  const int b = o / (3 * SPTS);
  out1[o] = new_xyz[((size_t)b * SPTS + s) * 3 + d];
}

// ---------------------------------------------------------------------------
extern "C" void kernel_launch(void* const* d_in, const int* in_sizes, int n_in,
                              void* d_out, int out_size, void* d_ws, size_t ws_size,
                              hipStream_t stream) {
  const float* xyz     = (const float*)d_in[0];
  const float* points  = (const float*)d_in[1];
  const float* conv0_w = (const float*)d_in[2];
  const float* conv0_b = (const float*)d_in[3];
  const float* bn0_g   = (const float*)d_in[4];
  const float* bn0_b   = (const float*)d_in[5];
  const float* conv1_w = (const float*)d_in[6];
  const float* conv1_b = (const float*)d_in[7];
  const float* bn1_g   = (const float*)d_in[8];
  const float* bn1_b   = (const float*)d_in[9];
  const float* conv2_w = (const float*)d_in[10];
  const float* conv2_b = (const float*)d_in[11];
  const float* bn2_g   = (const float*)d_in[12];
  const float* bn2_b   = (const float*)d_in[13];

  char* ws = (char*)d_ws;
  float* xyz_t   = (float*)(ws + OFF_XYZT);
  float* pts_t   = (float*)(ws + OFF_PTST);
  int*   cidx    = (int*)  (ws + OFF_CIDX);
  float* new_xyz = (float*)(ws + OFF_NEWX);
  int*   gidx    = (int*)  (ws + OFF_GIDX);
  float* feat    = (float*)(ws + OFF_FEAT);  // [M,68]; later reused as Y1 [M,64]
  float* y0      = (float*)(ws + OFF_Y0);
  float* y1      = feat;
  float* y2      = (float*)(ws + OFF_Y2);
  float* psum    = (float*)(ws + OFF_PART);
  float* psq     = psum + 256 * 128;
  float* scl     = (float*)(ws + OFF_SCL);
  float* shf     = (float*)(ws + OFF_SHF);

  float* out1 = (float*)d_out;                       // [B,3,S]
  float* out2 = out1 + (size_t)BATCH * 3 * SPTS;     // [B,128,S]

  // 0) transposes
  k_transpose<<<2048, 256, 0, stream>>>(xyz, points, xyz_t, pts_t);
  // 1) FPS
  k_fps<<<BATCH, 256, 0, stream>>>(xyz_t, cidx, new_xyz);
  // 2) kNN
  k_knn<<<BATCH * 4, 256, 0, stream>>>(xyz_t, new_xyz, gidx);
  // 3) feature build: 524288 waves
  k_build_feat<<<MROWS / 8, 256, 0, stream>>>(xyz_t, pts_t, new_xyz, gidx, feat);

  const int gemm_blocks = MROWS / 16 / 8;  // 4096

  // layer 0: 67(+pad)->64
  k_gemm_wmma<68, 67, 64><<<gemm_blocks, 256, 0, stream>>>(feat, conv0_w, conv0_b, y0);
  k_bn_partial<64><<<256, 256, 0, stream>>>(y0, psum, psq);
  k_bn_finalize<64><<<1, 256, 0, stream>>>(psum, psq, bn0_g, bn0_b, scl, shf);
  k_bn_apply<64><<<4096, 256, 0, stream>>>(y0, scl, shf);

  // layer 1: 64->64
  k_gemm_wmma<64, 64, 64><<<gemm_blocks, 256, 0, stream>>>(y0, conv1_w, conv1_b, y1);
  k_bn_partial<64><<<256, 256, 0, stream>>>(y1, psum, psq);
  k_bn_finalize<64><<<1, 256, 0, stream>>>(psum, psq, bn1_g, bn1_b, scl, shf);
  k_bn_apply<64><<<4096, 256, 0, stream>>>(y1, scl, shf);

  // layer 2: 64->128
  k_gemm_wmma<64, 64, 128><<<gemm_blocks, 256, 0, stream>>>(y1, conv2_w, conv2_b, y2);
  k_bn_partial<128><<<256, 256, 0, stream>>>(y2, psum, psq);
  k_bn_finalize<128><<<1, 256, 0, stream>>>(psum, psq, bn2_g, bn2_b, scl, shf);
  k_bn_apply<128><<<4096, 256, 0, stream>>>(y2, scl, shf);

  // outputs
  k_maxpool_tr<<<BATCH * SPTS / 8, 256, 0, stream>>>(y2, out2);
  k_newxyz_out<<<BATCH * 3 * SPTS / 256, 256, 0, stream>>>(new_xyz, out1);
}